// NNUEModel_40252433498261
// MI455X (gfx1250) — compile-verified
//
#include <hip/hip_runtime.h>
#include <hip/hip_bf16.h>
#include <math.h>

// ---- problem constants (from reference) ----
#define L1C   1024
#define L2C   15
#define L3C   32
#define NBK   8
#define NRFC  16
#define BSZ   4096
#define MF    32
#define OC    1032          // L1 + NB
#define OC4   258           // OC / 4
#define L0CORR (127.0f/128.0f)
#define TAUC  1.0f

typedef __attribute__((ext_vector_type(2))) float v2f;
typedef __attribute__((ext_vector_type(8))) float v8f;

static __device__ __forceinline__ float clamp01(float v) {
    return fminf(fmaxf(v, 0.0f), 1.0f);
}

// =====================================================================
// K0: rearrange l1_w [128 x 1024] into WMMA B-fragment-major layout.
// Fragment for (n_tile nt, k_step s, lane l): B[k][n] with n = nt*16 + (l%16),
// k = 4*s + 2*(l/16) + {0,1}.  One v2f per (nt,s,lane); linear index
// gid = nt*8192 + s*32 + lane  ->  per k-step load is a coalesced 256B/wave.
// =====================================================================
__global__ __launch_bounds__(256) void k0_prep_b(const float* __restrict__ l1w,
                                                 v2f* __restrict__ bfrag) {
    int gid  = blockIdx.x * blockDim.x + threadIdx.x;   // 0 .. 65535
    int lane = gid & 31;
    int s    = (gid >> 5) & 255;
    int nt   = gid >> 13;
    int n    = nt * 16 + (lane & 15);
    int k    = 4 * s + 2 * (lane >> 4);
    v2f v;
    v.x = l1w[(size_t)n * 1024 + k];
    v.y = l1w[(size_t)n * 1024 + k + 1];
    bfrag[gid] = v;
}

// =====================================================================
// K1: feature transformer. One workgroup per batch row.
// Thread t owns float4 column-group c4=t (cols 4t..4t+3); gathers 32 rows
// per side (rows are L2-resident), adds bias, does us/them mixing + clip,
// pairwise products p0/p1, writes l0_ (*L0CORR), router feats, psqt.
// =====================================================================
__global__ __launch_bounds__(256) void k1_ft(const float* __restrict__ us,
                                             const float* __restrict__ them,
                                             const float* __restrict__ wval,
                                             const float* __restrict__ bval,
                                             const float* __restrict__ ftw,
                                             const float* __restrict__ ftb,
                                             const int*   __restrict__ widx,
                                             const int*   __restrict__ bidx,
                                             float4* __restrict__ l0q,     // [B,256] float4
                                             float*  __restrict__ rf,      // [B,32]
                                             float4* __restrict__ wpsqt4,  // [B,2] float4
                                             float4* __restrict__ bpsqt4)  // [B,2] float4
{
    __shared__ int   siw[MF], sib[MF];
    __shared__ float svw[MF], svb[MF];
    __shared__ float4 Al[256], Bl[256];

    const int b = blockIdx.x;
    const int t = threadIdx.x;

    if (t < MF)                 { siw[t] = widx[b*MF + t];      svw[t] = wval[b*MF + t]; }
    else if (t < 2*MF)          { int m = t - MF;
                                  sib[m] = bidx[b*MF + m];      svb[m] = bval[b*MF + m]; }
    __syncthreads();

    const float ub = us[b], tb = them[b];
    const float4* __restrict__ ftw4 = (const float4*)ftw;
    const float4* __restrict__ ftb4 = (const float4*)ftb;

    // ---- main 1024 columns ----
    {
        float4 aw = ftb4[t];
        float4 ab = aw;
        for (int m = 0; m < MF; ++m) {
            const float4 rwv = ftw4[(size_t)siw[m] * OC4 + t];
            const float  vw  = svw[m];
            aw.x += vw * rwv.x; aw.y += vw * rwv.y; aw.z += vw * rwv.z; aw.w += vw * rwv.w;
            const float4 rbv = ftw4[(size_t)sib[m] * OC4 + t];
            const float  vb  = svb[m];
            ab.x += vb * rbv.x; ab.y += vb * rbv.y; ab.z += vb * rbv.z; ab.w += vb * rbv.w;
        }
        float4 a, bb;
        a.x  = clamp01(ub*aw.x + tb*ab.x);  a.y  = clamp01(ub*aw.y + tb*ab.y);
        a.z  = clamp01(ub*aw.z + tb*ab.z);  a.w  = clamp01(ub*aw.w + tb*ab.w);
        bb.x = clamp01(ub*ab.x + tb*aw.x);  bb.y = clamp01(ub*ab.y + tb*aw.y);
        bb.z = clamp01(ub*ab.z + tb*aw.z);  bb.w = clamp01(ub*ab.w + tb*aw.w);
        Al[t] = a; Bl[t] = bb;
    }

    // ---- psqt columns 1024..1031 (threads 0,1) ----
    if (t < 2) {
        const int c4 = 256 + t;
        float4 aw = ftb4[c4];
        float4 ab = aw;
        for (int m = 0; m < MF; ++m) {
            const float4 rwv = ftw4[(size_t)siw[m] * OC4 + c4];
            const float  vw  = svw[m];
            aw.x += vw * rwv.x; aw.y += vw * rwv.y; aw.z += vw * rwv.z; aw.w += vw * rwv.w;
            const float4 rbv = ftw4[(size_t)sib[m] * OC4 + c4];
            const float  vb  = svb[m];
            ab.x += vb * rbv.x; ab.y += vb * rbv.y; ab.z += vb * rbv.z; ab.w += vb * rbv.w;
        }
        wpsqt4[(size_t)b*2 + t] = aw;
        bpsqt4[(size_t)b*2 + t] = ab;
    }
    __syncthreads();

    // ---- pairwise products: p0[j]=l0[j]*l0[512+j]; p1[j]=l0[1024+j]*l0[1536+j] ----
    float4 p;
    if (t < 128) { const float4 x = Al[t],       y = Al[t + 128];
                   p.x = x.x*y.x; p.y = x.y*y.y; p.z = x.z*y.z; p.w = x.w*y.w; }
    else         { const float4 x = Bl[t - 128], y = Bl[t];
                   p.x = x.x*y.x; p.y = x.y*y.y; p.z = x.z*y.z; p.w = x.w*y.w; }

    float4 ps; ps.x = p.x*L0CORR; ps.y = p.y*L0CORR; ps.z = p.z*L0CORR; ps.w = p.w*L0CORR;
    l0q[(size_t)b * 256 + t] = ps;

    // router features: p0[496..511] (threads 124..127), p1[496..511] (threads 252..255), UNscaled
    if (t >= 124 && t < 128) {
        float* d = rf + (size_t)b*32 + (t - 124)*4;
        d[0] = p.x; d[1] = p.y; d[2] = p.z; d[3] = p.w;
    }
    if (t >= 252) {
        float* d = rf + (size_t)b*32 + 16 + (t - 252)*4;
        d[0] = p.x; d[1] = p.y; d[2] = p.z; d[3] = p.w;
    }
}

// =====================================================================
// K2: gumbel-softmax router. One thread per (row, bucket); groups of 8
// lanes per row, group-local reductions via __shfl_xor(width=8).
// rw = (hard - y_soft) + y_soft  (same FP order as reference).
// =====================================================================
__global__ __launch_bounds__(256) void k2_router(const float* __restrict__ rf,
                                                 const float* __restrict__ rww,
                                                 const float* __restrict__ rwb,
                                                 const float* __restrict__ rls,
                                                 const float* __restrict__ gumbel,
                                                 float* __restrict__ rwout) {
    const int gid = blockIdx.x * blockDim.x + threadIdx.x;  // B*8
    const int b = gid >> 3, n = gid & 7;

    float dot = rwb[n];
    #pragma unroll
    for (int i = 0; i < 32; ++i) dot += rf[(size_t)b*32 + i] * rww[n*32 + i];
    const float logit = rls[0] * dot;
    const float z = (logit + gumbel[(size_t)b*8 + n]) * (1.0f / TAUC);

    float m = z;
    m = fmaxf(m, __shfl_xor(m, 1, 8));
    m = fmaxf(m, __shfl_xor(m, 2, 8));
    m = fmaxf(m, __shfl_xor(m, 4, 8));
    const float e = expf(z - m);
    float sum = e;
    sum += __shfl_xor(sum, 1, 8);
    sum += __shfl_xor(sum, 2, 8);
    sum += __shfl_xor(sum, 4, 8);
    const float y = e / sum;

    int cand = (z == m) ? n : 8;      // first index attaining the max
    cand = min(cand, __shfl_xor(cand, 1, 8));
    cand = min(cand, __shfl_xor(cand, 2, 8));
    cand = min(cand, __shfl_xor(cand, 4, 8));
    const float hard = (cand == n) ? 1.0f : 0.0f;

    rwout[gid] = (hard - y) + y;
}

// =====================================================================
// K3: L1 GEMM  [4096x1024] x [1024x128]  via V_WMMA_F32_16X16X4_F32.
// One WG (8 waves) per 16-row tile; wave w owns columns [16w,16w+16)
// == expert bucket w.  A tile staged in LDS (stride 1028 -> conflict-free
// fragment reads); B fragments streamed coalesced from K0's buffer.
// Epilogue: +bias, *rw[b][w], ds_add_f32 across waves, clip/square -> l1x,l1y.
// =====================================================================
#define AS_STRIDE 1028
__global__ __launch_bounds__(256) void k3_l1_wmma(const float4* __restrict__ l0q,
                                                  const v2f*    __restrict__ bfrag,
                                                  const float*  __restrict__ l1b,
                                                  const float*  __restrict__ rw,
                                                  float* __restrict__ l1x,   // [B,30]
                                                  float* __restrict__ l1y)   // [B]
{
    extern __shared__ float smem[];
    float* As = smem;                      // 16 * 1028 floats
    float* Rs = smem + 16 * AS_STRIDE;     // 16 rows x 8 buckets
    float* Ls = Rs + 128;                  // 16 rows x 16 outputs

    const int t        = threadIdx.x;
    const int row_base = blockIdx.x * 16;

    // stage A tile: 16 rows x 256 float4
    #pragma unroll
    for (int it = 0; it < 16; ++it) {
        const int idx = t + 256 * it;          // 0..4095
        const int r   = idx >> 8;
        const int c4  = idx & 255;
        const float4 v = l0q[(size_t)(row_base + r) * 256 + c4];
        float* d = &As[r * AS_STRIDE + 4 * c4];
        d[0] = v.x; d[1] = v.y; d[2] = v.z; d[3] = v.w;
    }
    if (t < 128) Rs[t] = rw[(size_t)(row_base + (t >> 3)) * 8 + (t & 7)];
    Ls[t] = 0.0f;
    __syncthreads();

    const int lane = t & 31;
    const int w    = t >> 5;
    const int nlo  = lane & 15;
    const int hi   = lane >> 4;

    const float* arow = &As[nlo * AS_STRIDE + 2 * hi];
    const v2f*   bp   = bfrag + (size_t)w * 8192 + lane;

    v8f acc = {0.f, 0.f, 0.f, 0.f, 0.f, 0.f, 0.f, 0.f};
    for (int s = 0; s < 256; ++s) {
        const v2f a  = *(const v2f*)(arow + 4 * s);   // ds_load_b64, bank-conflict-free
        const v2f bv = bp[(size_t)s * 32];            // global_load_b64, 256B/wave coalesced
        acc = __builtin_amdgcn_wmma_f32_16x16x4_f32(
                  false, a, false, bv, (short)0, acc, false, false);
    }

    const float bias = l1b[w * 16 + nlo];
    #pragma unroll
    for (int v = 0; v < 8; ++v) {
        const int rloc = v + 8 * hi;                      // C layout: VGPR v -> rows v, v+8
        const float scaled = (acc[v] + bias) * Rs[rloc * 8 + w];
        atomicAdd(&Ls[rloc * 16 + nlo], scaled);          // ds_add_f32
    }
    __syncthreads();

    {
        const int r = t >> 4, j = t & 15;
        const float val = Ls[t];
        if (j < 15) {
            const float c = clamp01(val);
            l1x[(size_t)(row_base + r) * 30 + j]      = c * c * L0CORR;
            l1x[(size_t)(row_base + r) * 30 + 15 + j] = c;
        } else {
            l1y[row_base + r] = val;
        }
    }
}

// =====================================================================
// K4: L2 + output layers + psqt.  One wave per batch row (8 rows / WG).
// l2_w staged in LDS; lane j owns output column j across all 8 buckets;
// L3 dot via full-wave __shfl_xor butterfly.
// =====================================================================
__global__ __launch_bounds__(256) void k4_tail(const float* __restrict__ l1x,
                                               const float* __restrict__ l1y,
                                               const float* __restrict__ rw,
                                               const float* __restrict__ l2w,
                                               const float* __restrict__ l2b,
                                               const float* __restrict__ ow,
                                               const float* __restrict__ ob,
                                               const float* __restrict__ wpsqt,
                                               const float* __restrict__ bpsqt,
                                               const float* __restrict__ us,
                                               float* __restrict__ out) {
    __shared__ float sw[256 * 30];
    __shared__ float sb[256];
    __shared__ float sow[256];
    __shared__ float sob[8];

    const int t = threadIdx.x;
    for (int i = t; i < 256 * 30; i += 256) sw[i] = l2w[i];
    sb[t]  = l2b[t];
    sow[t] = ow[t];
    if (t < 8) sob[t] = ob[t];
    __syncthreads();

    const int wv = t >> 5, j = t & 31;
    const int b  = blockIdx.x * 8 + wv;

    float x[30];
    #pragma unroll
    for (int i = 0; i < 30; ++i) x[i] = l1x[(size_t)b * 30 + i];
    float r[8];
    #pragma unroll
    for (int n = 0; n < 8; ++n) r[n] = rw[(size_t)b * 8 + n];

    float l2c = 0.0f;
    #pragma unroll
    for (int n = 0; n < 8; ++n) {
        const int o = n * 32 + j;
        float s = sb[o];
        const float* wr = &sw[o * 30];
        #pragma unroll
        for (int i = 0; i < 30; ++i) s += x[i] * wr[i];
        l2c += r[n] * s;
    }
    l2c = clamp01(l2c);

    float res = 0.0f;
    #pragma unroll
    for (int n = 0; n < 8; ++n) {
        float v = sow[n * 32 + j] * l2c;
        v += __shfl_xor(v, 16, 32);
        v += __shfl_xor(v,  8, 32);
        v += __shfl_xor(v,  4, 32);
        v += __shfl_xor(v,  2, 32);
        v += __shfl_xor(v,  1, 32);
        res += r[n] * (v + sob[n]);
    }

    if (j == 0) {
        float wps = 0.0f, bps = 0.0f;
        #pragma unroll
        for (int n = 0; n < 8; ++n) {
            wps += r[n] * wpsqt[(size_t)b * 8 + n];
            bps += r[n] * bpsqt[(size_t)b * 8 + n];
        }
        out[b] = res + l1y[b] + (wps - bps) * (us[b] - 0.5f);
    }
}

// =====================================================================
extern "C" void kernel_launch(void* const* d_in, const int* in_sizes, int n_in,
                              void* d_out, int out_size, void* d_ws, size_t ws_size,
                              hipStream_t stream) {
    const float* us     = (const float*)d_in[0];
    const float* them   = (const float*)d_in[1];
    const float* wval   = (const float*)d_in[2];
    const float* bval   = (const float*)d_in[3];
    const float* gumbel = (const float*)d_in[4];
    const float* ftw    = (const float*)d_in[5];
    const float* ftb    = (const float*)d_in[6];
    const float* rww    = (const float*)d_in[7];
    const float* rwb    = (const float*)d_in[8];
    const float* rls    = (const float*)d_in[9];
    const float* l1w    = (const float*)d_in[10];
    const float* l1b    = (const float*)d_in[11];
    const float* l2w    = (const float*)d_in[12];
    const float* l2b    = (const float*)d_in[13];
    const float* ow     = (const float*)d_in[14];
    const float* ob     = (const float*)d_in[15];
    const int*   widx   = (const int*)d_in[16];
    const int*   bidx   = (const int*)d_in[17];
    float* out = (float*)d_out;

    char* ws = (char*)d_ws;
    size_t off = 0;
    auto alloc = [&](size_t bytes) {
        void* p = ws + off;
        off += (bytes + 255) & ~(size_t)255;
        return p;
    };
    float*  l0_    = (float*) alloc((size_t)BSZ * 1024 * 4);  // 16 MB
    v2f*    bfrag  = (v2f*)   alloc((size_t)65536 * 8);       // 512 KB
    float*  rf     = (float*) alloc((size_t)BSZ * 32 * 4);
    float4* wpsqt4 = (float4*)alloc((size_t)BSZ * 8 * 4);
    float4* bpsqt4 = (float4*)alloc((size_t)BSZ * 8 * 4);
    float*  rwbuf  = (float*) alloc((size_t)BSZ * 8 * 4);
    float*  l1x    = (float*) alloc((size_t)BSZ * 30 * 4);
    float*  l1y    = (float*) alloc((size_t)BSZ * 4);

    k0_prep_b<<<256, 256, 0, stream>>>(l1w, bfrag);

    k1_ft<<<BSZ, 256, 0, stream>>>(us, them, wval, bval, ftw, ftb, widx, bidx,
                                   (float4*)l0_, rf, wpsqt4, bpsqt4);

    k2_router<<<(BSZ * NBK) / 256, 256, 0, stream>>>(rf, rww, rwb, rls, gumbel, rwbuf);

    const size_t k3_lds = (size_t)(16 * AS_STRIDE + 128 + 256) * sizeof(float);
    k3_l1_wmma<<<BSZ / 16, 256, k3_lds, stream>>>((const float4*)l0_, bfrag, l1b,
                                                  rwbuf, l1x, l1y);

    k4_tail<<<BSZ / 8, 256, 0, stream>>>(l1x, l1y, rwbuf, l2w, l2b, ow, ob,
                                         (const float*)wpsqt4, (const float*)bpsqt4,
                                         us, out);
}